// Capsule_18098992185786
// MI455X (gfx1250) — compile-verified
//
#include <hip/hip_runtime.h>
#include <hip/hip_bf16.h>

typedef __attribute__((ext_vector_type(2))) float v2f;
typedef __attribute__((ext_vector_type(4))) float v4f;
typedef __attribute__((ext_vector_type(8))) float v8f;

#define CAP_B 256
#define CAP_I 1152
#define CAP_O 10
#define CAP_DI 8
#define CAP_DO 16
#define CAP_R 3

// -----------------------------------------------------------------------------
// Phase 1: hat_u[b,i,o,do] = sum_k W[i,o,do,k] * x[b,i,k]
// One wave32 per (b_tile of 16, i); loops over all O=10 output capsules,
// reusing the A-matrix (x tile) in registers. Each o: D[16x16] = X[16x8] @
// W^T[8x16] as two accumulated V_WMMA_F32_16X16X4_F32 (K = 0..3, then 4..7).
// -----------------------------------------------------------------------------
__global__ __launch_bounds__(256) void hatu_wmma_kernel(
    const float* __restrict__ x,     // (B, I, 8)
    const float* __restrict__ w,     // (I*O, 16, 8)
    float* __restrict__ hat)         // (B, I, O, 16)
{
    const int gwid = blockIdx.x * 8 + (threadIdx.x >> 5);   // 0 .. 16*1152-1
    const int lane = threadIdx.x & 31;

    const int i  = gwid % CAP_I;
    const int bt = gwid / CAP_I;                            // 0..15 (batch tile)

    const int n     = lane & 15;        // A: row (batch-in-tile); B: col (do)
    const int khalf = lane >> 4;        // 0 or 1
    const int kb    = khalf * 2;        // K base: 0 or 2 (then +4 for hi half)

    // A: x row for batch bt*16+n, capsule i  (loaded once, reused for all o)
    const float* xrow = x + ((size_t)(bt * 16 + n) * CAP_I + i) * CAP_DI;
    const v2f alo = *(const v2f*)(xrow + kb);
    const v2f ahi = *(const v2f*)(xrow + kb + 4);

    // B base: W[i, o, do=n, k];  o-stride = 16*8 floats
    const float* wbase = w + ((size_t)i * CAP_O * CAP_DO + n) * CAP_DI;

    const size_t bstride = (size_t)CAP_I * CAP_O * CAP_DO;  // per-batch stride
    float* dbase = hat + ((size_t)(bt * 16) * CAP_I + i) * CAP_O * CAP_DO + n;
    const int mofs = khalf * 8;

#pragma unroll
    for (int o = 0; o < CAP_O; ++o) {
        const float* wrow = wbase + (size_t)o * CAP_DO * CAP_DI;
        const v2f blo = *(const v2f*)(wrow + kb);
        const v2f bhi = *(const v2f*)(wrow + kb + 4);

        v8f acc = {};
        acc = __builtin_amdgcn_wmma_f32_16x16x4_f32(
            false, alo, false, blo, (short)0, acc, false, false);
        acc = __builtin_amdgcn_wmma_f32_16x16x4_f32(
            false, ahi, false, bhi, (short)0, acc, false, false);

        // D layout: VGPR j -> M = j + 8*khalf (batch row), N = lane&15 (do)
        float* dst = dbase + o * CAP_DO;
#pragma unroll
        for (int j = 0; j < 8; ++j)
            dst[(size_t)(j + mofs) * bstride] = acc[j];
    }
}

// -----------------------------------------------------------------------------
// Phase 2: dynamic routing, one block per batch element.
// Logits b[i,o] live in registers (thread t owns i = t and t+640).
// LDS holds c (coupling coeffs), reduction scratch, s and v.
// -----------------------------------------------------------------------------
__global__ __launch_bounds__(640) void routing_kernel(
    const float* __restrict__ hat,   // (B, I, O, 16)
    float* __restrict__ out)         // (B, O, 16)
{
    const int b = blockIdx.x;
    const int t = threadIdx.x;

    __shared__ float cc[CAP_I * CAP_O];                 // 46080 B
    __shared__ __align__(16) float red[4 * CAP_O * CAP_DO];
    __shared__ __align__(16) float sarr[CAP_O * CAP_DO];
    __shared__ __align__(16) float varr[CAP_O * CAP_DO];

    const float* hb = hat + (size_t)b * CAP_I * CAP_O * CAP_DO;

    float bf[2][CAP_O];
#pragma unroll
    for (int rep = 0; rep < 2; ++rep)
#pragma unroll
        for (int o = 0; o < CAP_O; ++o) bf[rep][o] = 0.0f;

    for (int r = 0; r < CAP_R; ++r) {
        // ---- sub-pass A: agreement update (r>0) + softmax -> cc ----
        for (int rep = 0; rep < 2; ++rep) {
            const int i = t + rep * 640;
            if (i < CAP_I) {
                if (r > 0) {
#pragma unroll
                    for (int o = 0; o < CAP_O; ++o) {
                        const float* hrow = hb + ((size_t)i * CAP_O + o) * CAP_DO;
                        float dot = 0.0f;
#pragma unroll
                        for (int q = 0; q < 4; ++q) {
                            v4f h = *(const v4f*)(hrow + q * 4);
                            v4f v = *(const v4f*)(&varr[o * CAP_DO + q * 4]);
                            dot += h.x * v.x + h.y * v.y + h.z * v.z + h.w * v.w;
                        }
                        bf[rep][o] += dot;
                    }
                }
                float mx = bf[rep][0];
#pragma unroll
                for (int o = 1; o < CAP_O; ++o) mx = fmaxf(mx, bf[rep][o]);
                float e[CAP_O];
                float sum = 0.0f;
#pragma unroll
                for (int o = 0; o < CAP_O; ++o) {
                    e[o] = __expf(bf[rep][o] - mx);
                    sum += e[o];
                }
                const float inv = 1.0f / sum;
#pragma unroll
                for (int o = 0; o < CAP_O; ++o)
                    cc[i * CAP_O + o] = e[o] * inv;
            }
        }
        __syncthreads();

        // ---- sub-pass B: s[o,do] = sum_i c[i,o] * hat_u[i,o,do] ----
        {
            const int idx   = t % 160;          // o*16 + do
            const int chunk = t / 160;          // 0..3, 288 capsules each
            const int o = idx >> 4;
            const int d = idx & 15;
            float part = 0.0f;
            const int i0 = chunk * 288;
            const int i1 = i0 + 288;
            for (int i = i0; i < i1; ++i)
                part += cc[i * CAP_O + o] * hb[((size_t)i * CAP_O + o) * CAP_DO + d];
            red[chunk * 160 + idx] = part;
        }
        __syncthreads();

        if (t < 160)
            sarr[t] = red[t] + red[160 + t] + red[320 + t] + red[480 + t];
        __syncthreads();

        // ---- squash: v = s * ||s|| / (1 + ||s||^2) ----
        if (t < 160) {
            const int o = t >> 4;
            float n2 = 0.0f;
#pragma unroll
            for (int d = 0; d < CAP_DO; ++d) {
                const float sv = sarr[o * CAP_DO + d];
                n2 += sv * sv;
            }
            const float scale = sqrtf(n2) / (n2 + 1.0f);
            const float v = sarr[t] * scale;
            varr[t] = v;
            if (r == CAP_R - 1)
                out[(size_t)b * 160 + t] = v;
        }
        __syncthreads();
    }
}

extern "C" void kernel_launch(void* const* d_in, const int* in_sizes, int n_in,
                              void* d_out, int out_size, void* d_ws, size_t ws_size,
                              hipStream_t stream) {
    const float* x = (const float*)d_in[0];   // (256, 1152, 8) f32
    const float* w = (const float*)d_in[1];   // (11520, 16, 8) f32
    float* out = (float*)d_out;               // (256, 10, 16) f32
    float* hat = (float*)d_ws;                // 189 MB scratch: (256,1152,10,16) f32

    // 16 batch tiles * 1152 wave-tiles (each wave covers all 10 o's),
    // 8 waves (256 threads) per block
    const int total_waves = 16 * CAP_I;               // 18432
    hatu_wmma_kernel<<<dim3(total_waves / 8), dim3(256), 0, stream>>>(x, w, hat);
    routing_kernel<<<dim3(CAP_B), dim3(640), 0, stream>>>(hat, out);
}